// Rocket_48661979464269
// MI455X (gfx1250) — compile-verified
//
#include <hip/hip_runtime.h>
#include <hip/hip_bf16.h>

// Problem constants (match the reference)
#define KNUM   5000
#define BROWS  32
#define TLEN   512
#define KLEN   9
#define DMODEL 10000   // 2*KNUM
#define PREDN  96
#define NGRP   64      // (dil 1..32) x (pad flag 0/1)
#define MAXT   384     // >= sum_g ceil(cnt_g/16)  (<= 5000/16 + 64 = 377)
#define NBLK   ((KNUM + 255) / 256)   // 20 histogram blocks

typedef __attribute__((ext_vector_type(2))) float v2f;
typedef __attribute__((ext_vector_type(8))) float v8f;

__device__ __forceinline__ int group_of(int d, int p) {
    return (d - 1) * 2 + (p > 0 ? 1 : 0);
}

__device__ __forceinline__ void wait_async_zero() {
#if __has_builtin(__builtin_amdgcn_s_wait_asynccnt)
    __builtin_amdgcn_s_wait_asynccnt(0);
#else
    asm volatile("s_wait_asynccnt 0" ::: "memory");
#endif
}

// ---------------------------------------------------------------------------
// Bucketing step 1: per-launch-block histogram of (dil,pad) groups.
// Deterministic: shared atomicAdd counts are order-independent.
// ---------------------------------------------------------------------------
__global__ __launch_bounds__(256) void k_hist(
    const int* __restrict__ dil, const int* __restrict__ pad,
    int* __restrict__ blockCnt)          // [NBLK][NGRP]
{
    __shared__ int sh[NGRP];
    if (threadIdx.x < NGRP) sh[threadIdx.x] = 0;
    __syncthreads();
    int k = blockIdx.x * 256 + threadIdx.x;
    if (k < KNUM) atomicAdd(&sh[group_of(dil[k], pad[k])], 1);
    __syncthreads();
    if (threadIdx.x < NGRP)
        blockCnt[blockIdx.x * NGRP + threadIdx.x] = sh[threadIdx.x];
}

// ---------------------------------------------------------------------------
// Bucketing step 2: serial block-prefix per group + group offsets + tile
// table (64*20 trivial iterations). tileCnt[] beyond live tiles stays 0.
// ---------------------------------------------------------------------------
__global__ void k_scan(
    const int* __restrict__ blockCnt, int* __restrict__ blockBase,
    int* __restrict__ grpOff,
    int* __restrict__ tileG, int* __restrict__ tileStart, int* __restrict__ tileCnt)
{
    if (threadIdx.x != 0 || blockIdx.x != 0) return;
    int off = 0, tb = 0;
    for (int g = 0; g < NGRP; ++g) {
        int c = 0;
        for (int bk = 0; bk < NBLK; ++bk) {
            blockBase[bk * NGRP + g] = c;
            c += blockCnt[bk * NGRP + g];
        }
        grpOff[g] = off;
        int nt = (c + 15) >> 4;
        for (int j = 0; j < nt; ++j) {
            tileG[tb]     = g;
            tileStart[tb] = off + j * 16;
            tileCnt[tb]   = (c - j * 16 < 16) ? (c - j * 16) : 16;
            ++tb;
        }
        off += c;
    }
}

// ---------------------------------------------------------------------------
// Bucketing step 3: deterministic scatter. Local rank from LDS (<=255 reads),
// global position = grpOff + blockBase + localRank.
// ---------------------------------------------------------------------------
__global__ __launch_bounds__(256) void k_rank(
    const int* __restrict__ dil, const int* __restrict__ pad,
    const int* __restrict__ grpOff, const int* __restrict__ blockBase,
    int* __restrict__ sortedIdx)
{
    __shared__ int gsh[256];
    int k = blockIdx.x * 256 + threadIdx.x;
    int g = -1;
    if (k < KNUM) g = group_of(dil[k], pad[k]);
    gsh[threadIdx.x] = g;
    __syncthreads();
    if (k >= KNUM) return;
    int local = 0;
    for (int j = 0; j < threadIdx.x; ++j) local += (gsh[j] == g) ? 1 : 0;
    sortedIdx[grpOff[g] + blockBase[blockIdx.x * NGRP + g] + local] = k;
}

// ---------------------------------------------------------------------------
// ROCKET conv via fp32 WMMA. Grid (MAXT, 4): block = one 16-kernel tile x
// 8 batch rows; wave w handles batch b = blockIdx.y*8 + w. All kernels in a
// tile share (dil,pad) => shared B matrix and uniform out_len.
//   A(16x12) = padded weights (taps 9..11 = 0), three K=4 chunks
//   B(12x16)[j,n] = x[b, t0+n - pad + j*dil]  (0 outside [0,T))
// x rows staged with async LDS loads (global_load_async_to_lds_b128,
// ASYNCcnt) then s_wait_asynccnt 0 + barrier.
// Fragment layout per CDNA5 ISA: lanes 0-15 K={0,1}, lanes 16-31 K={2,3};
// C/D: VGPR r -> M=r (lanes 0-15) / M=r+8 (lanes 16-31), N=lane&15.
// ---------------------------------------------------------------------------
__global__ __launch_bounds__(256) void k_conv_wmma(
    const float* __restrict__ x,        // [B,T]
    const float* __restrict__ w,        // [K,9]
    const float* __restrict__ bias,     // [K]
    const int*   __restrict__ tileG,
    const int*   __restrict__ tileStart,
    const int*   __restrict__ tileCnt,
    const int*   __restrict__ sortedIdx,
    float*       __restrict__ feat)     // [B, 2K]
{
    __shared__ __align__(16) float xs[8 * TLEN];   // 16 KB (8 batch rows)
    __shared__ float wmat[16][12];
    __shared__ float bsh[16];
    __shared__ int   kidx[16];

    const int tt   = blockIdx.x;
    const int cntK = tileCnt[tt];
    if (cntK == 0) return;              // uniform across block (no barrier yet)

    const int g     = tileG[tt];
    const int start = tileStart[tt];
    const int d     = (g >> 1) + 1;
    const int p     = (g & 1) ? 4 * d : 0;              // ((KLEN-1)*d/2)*flag
    const int out_len = TLEN + 2 * p - (KLEN - 1) * d;  // uniform per tile

    // Async-stage this block's 8 batch rows of x into LDS (ASYNCcnt path).
    const float* __restrict__ xg = x + (size_t)blockIdx.y * 8 * TLEN;
    for (int i = threadIdx.x * 4; i < 8 * TLEN; i += 256 * 4) {
        unsigned     ldsa = (unsigned)(uintptr_t)(xs + i);  // LDS byte address
        const float* ga   = xg + i;
        asm volatile("global_load_async_to_lds_b128 %0, %1, off"
                     :: "v"(ldsa), "v"(ga) : "memory");
    }

    if (threadIdx.x < 16) {
        int m  = threadIdx.x;
        int kk = (m < cntK) ? sortedIdx[start + m] : -1;
        kidx[m] = kk;
        bsh[m]  = (kk >= 0) ? bias[kk] : 0.0f;
#pragma unroll
        for (int j = 0; j < 12; ++j)
            wmat[m][j] = (kk >= 0 && j < KLEN) ? w[kk * KLEN + j] : 0.0f;
    }
    wait_async_zero();
    __syncthreads();

    const int lane = threadIdx.x & 31;
    const int wave = threadIdx.x >> 5;
    const int half = lane >> 4;
    const int l    = lane & 15;
    const int kb   = half * 2;          // K base inside each 4-chunk

    // Loop-invariant A fragments (taps [0..3], [4..7], [8..11])
    v2f a0, a1, a2;
    a0.x = wmat[l][kb];     a0.y = wmat[l][kb + 1];
    a1.x = wmat[l][4 + kb]; a1.y = wmat[l][4 + kb + 1];
    a2.x = wmat[l][8 + kb]; a2.y = wmat[l][8 + kb + 1];

    float brow[8];
#pragma unroll
    for (int r = 0; r < 8; ++r) brow[r] = bsh[half * 8 + r];

    const int nfull = out_len >> 4;
    const int tail  = out_len & 15;
    const int ntile = nfull + (tail ? 1 : 0);
    const v8f czero = {};

    const int b = blockIdx.y * 8 + wave;            // this wave's batch row
    const float* __restrict__ xb = xs + wave * TLEN;

    int   cnt[8];
    float mx[8];
#pragma unroll
    for (int r = 0; r < 8; ++r) { cnt[r] = 0; mx[r] = -__builtin_inff(); }

    auto gather = [&](int i) -> float {
        float v = xb[i & (TLEN - 1)];               // safe clamp
        return ((unsigned)i < (unsigned)TLEN) ? v : 0.0f;
    };

#pragma unroll 2
    for (int ti = 0; ti < ntile; ++ti) {
        const int t0   = ti << 4;
        const int base = t0 + l - p;
        const int j0   = base + kb * d;             // tap kb
        v2f b0, b1, b2;
        b0.x = gather(j0);           b0.y = gather(j0 + d);
        b1.x = gather(j0 + 4 * d);   b1.y = gather(j0 + 5 * d);
        b2.x = gather(j0 + 8 * d);   b2.y = gather(j0 + 9 * d);

        v8f c;
        c = __builtin_amdgcn_wmma_f32_16x16x4_f32(false, a0, false, b0,
                                                  (short)0, czero, false, false);
        c = __builtin_amdgcn_wmma_f32_16x16x4_f32(false, a1, false, b1,
                                                  (short)0, c, false, false);
        c = __builtin_amdgcn_wmma_f32_16x16x4_f32(false, a2, false, b2,
                                                  (short)0, c, false, false);

        const bool validN = (ti < nfull) | (l < tail);   // N mask (tail tile)
#pragma unroll
        for (int r = 0; r < 8; ++r) {
            float v = c[r] + brow[r];
            if (validN) {
                cnt[r] += (v > 0.0f) ? 1 : 0;
                mx[r]   = fmaxf(mx[r], v);
            }
        }
    }

    // Reduce across the 16 time-lanes inside each half (halves hold distinct
    // kernel rows M=r vs M=r+8). XOR masks 1..8 stay inside each half.
#pragma unroll
    for (int r = 0; r < 8; ++r) {
#pragma unroll
        for (int m = 8; m >= 1; m >>= 1) {
            cnt[r] += __shfl_xor(cnt[r], m, 32);
            mx[r]   = fmaxf(mx[r], __shfl_xor(mx[r], m, 32));
        }
    }

    if (l == 0) {
        const float inv = 1.0f / (float)out_len;
#pragma unroll
        for (int r = 0; r < 8; ++r) {
            int kk = kidx[half * 8 + r];
            if (kk >= 0) {
                feat[b * DMODEL + 2 * kk]     = (float)cnt[r] * inv;
                feat[b * DMODEL + 2 * kk + 1] = mx[r];
            }
        }
    }
}

// ---------------------------------------------------------------------------
// Per-row mean and rsqrt(var+eps) over DMODEL features.
// ---------------------------------------------------------------------------
__global__ __launch_bounds__(256) void k_stats(
    const float* __restrict__ feat, float* __restrict__ mu, float* __restrict__ rsig)
{
    __shared__ float sh[512];
    const int b = blockIdx.x;
    float s = 0.0f, sq = 0.0f;
    for (int f = threadIdx.x; f < DMODEL; f += 256) {
        float v = feat[b * DMODEL + f];
        s += v;
        sq = fmaf(v, v, sq);
    }
    sh[threadIdx.x]       = s;
    sh[256 + threadIdx.x] = sq;
    __syncthreads();
    for (int off = 128; off > 0; off >>= 1) {
        if (threadIdx.x < off) {
            sh[threadIdx.x]       += sh[threadIdx.x + off];
            sh[256 + threadIdx.x] += sh[256 + threadIdx.x + off];
        }
        __syncthreads();
    }
    if (threadIdx.x == 0) {
        float m   = sh[0]   * (1.0f / (float)DMODEL);
        float var = sh[256] * (1.0f / (float)DMODEL) - m * m;
        mu[b]   = m;
        rsig[b] = rsqrtf(var + 1e-5f);
    }
}

// ---------------------------------------------------------------------------
// Fold LayerNorm affine into head constants: S1=sum(gamma*W), S2=sum(beta*W).
// ---------------------------------------------------------------------------
__global__ __launch_bounds__(256) void k_proj(
    const float* __restrict__ gamma, const float* __restrict__ beta,
    const float* __restrict__ hw, float* __restrict__ S1, float* __restrict__ S2)
{
    __shared__ float sh[512];
    const int pp = blockIdx.x;
    float a = 0.0f, c = 0.0f;
    for (int f = threadIdx.x; f < DMODEL; f += 256) {
        float h = hw[pp * DMODEL + f];
        a = fmaf(gamma[f], h, a);
        c = fmaf(beta[f],  h, c);
    }
    sh[threadIdx.x]       = a;
    sh[256 + threadIdx.x] = c;
    __syncthreads();
    for (int off = 128; off > 0; off >>= 1) {
        if (threadIdx.x < off) {
            sh[threadIdx.x]       += sh[threadIdx.x + off];
            sh[256 + threadIdx.x] += sh[256 + threadIdx.x + off];
        }
        __syncthreads();
    }
    if (threadIdx.x == 0) { S1[pp] = sh[0]; S2[pp] = sh[256]; }
}

// ---------------------------------------------------------------------------
// Head GEMM via fp32 WMMA: G[b,p] = sum_f feat*gamma*hw; LayerNorm folded in.
// ---------------------------------------------------------------------------
__global__ __launch_bounds__(32) void k_head(
    const float* __restrict__ feat, const float* __restrict__ gamma,
    const float* __restrict__ hw,   const float* __restrict__ hb,
    const float* __restrict__ S1,   const float* __restrict__ S2,
    const float* __restrict__ mu,   const float* __restrict__ rsig,
    float*       __restrict__ out)
{
    const int nt = blockIdx.x % 6;
    const int mt = blockIdx.x / 6;
    const int b0 = mt * 16, p0 = nt * 16;

    const int lane = threadIdx.x;
    const int half = lane >> 4;
    const int l    = lane & 15;
    const int kb   = half * 2;

    const float* __restrict__ arow = feat  + (size_t)(b0 + l) * DMODEL + kb;
    const float* __restrict__ brow = hw    + (size_t)(p0 + l) * DMODEL + kb;
    const float* __restrict__ grow = gamma + kb;

    v8f c = {};
#pragma unroll 4
    for (int f0 = 0; f0 < DMODEL; f0 += 4) {
        if ((f0 & 255) == 0) {
            __builtin_prefetch(arow + f0 + 1024, 0, 1);
            __builtin_prefetch(brow + f0 + 1024, 0, 1);
        }
        v2f a  = *(const v2f*)(arow + f0);
        v2f gg = *(const v2f*)(grow + f0);
        v2f hh = *(const v2f*)(brow + f0);
        v2f bf = gg * hh;
        c = __builtin_amdgcn_wmma_f32_16x16x4_f32(false, a, false, bf,
                                                  (short)0, c, false, false);
    }

    const int pp = p0 + l;
    const float s1 = S1[pp], s2 = S2[pp], hbp = hb[pp];
#pragma unroll
    for (int r = 0; r < 8; ++r) {
        const int b = b0 + r + half * 8;
        out[b * PREDN + pp] = rsig[b] * (c[r] - mu[b] * s1) + s2 + hbp;
    }
}

// ---------------------------------------------------------------------------
// Host-side launcher
// ---------------------------------------------------------------------------
extern "C" void kernel_launch(void* const* d_in, const int* in_sizes, int n_in,
                              void* d_out, int out_size, void* d_ws, size_t ws_size,
                              hipStream_t stream) {
    const float* x     = (const float*)d_in[0];
    const float* w     = (const float*)d_in[1];
    const float* bias  = (const float*)d_in[2];
    const int*   dil   = (const int*)  d_in[3];
    const int*   pad   = (const int*)  d_in[4];
    const float* gamma = (const float*)d_in[5];
    const float* beta  = (const float*)d_in[6];
    const float* hw    = (const float*)d_in[7];
    const float* hb    = (const float*)d_in[8];
    float* out = (float*)d_out;

    // Workspace: feat[B*D] | mu | rsig | S1 | S2 | int meta[...]
    float* feat = (float*)d_ws;
    float* mu   = feat + (size_t)BROWS * DMODEL;
    float* rsig = mu   + BROWS;
    float* S1   = rsig + BROWS;
    float* S2   = S1   + PREDN;
    int*   meta      = (int*)(S2 + PREDN);
    int*   grpOff    = meta;
    int*   tileG     = grpOff + NGRP;
    int*   tileStart = tileG + MAXT;
    int*   tileCnt   = tileStart + MAXT;
    int*   sortedIdx = tileCnt + MAXT;
    int*   blockCnt  = sortedIdx + KNUM;
    int*   blockBase = blockCnt + NBLK * NGRP;

    const size_t metaBytes =
        (size_t)(NGRP + 3 * MAXT + KNUM + 2 * NBLK * NGRP) * sizeof(int);
    hipMemsetAsync(meta, 0, metaBytes, stream);

    k_hist<<<NBLK, 256, 0, stream>>>(dil, pad, blockCnt);
    k_scan<<<1, 64, 0, stream>>>(blockCnt, blockBase, grpOff,
                                 tileG, tileStart, tileCnt);
    k_rank<<<NBLK, 256, 0, stream>>>(dil, pad, grpOff, blockBase, sortedIdx);
    k_conv_wmma<<<dim3(MAXT, 4), 256, 0, stream>>>(x, w, bias, tileG, tileStart,
                                                   tileCnt, sortedIdx, feat);
    k_stats<<<BROWS, 256, 0, stream>>>(feat, mu, rsig);
    k_proj <<<PREDN, 256, 0, stream>>>(gamma, beta, hw, S1, S2);
    k_head <<<12, 32, 0, stream>>>(feat, gamma, hw, hb, S1, S2, mu, rsig, out);
}